// Teacher_3118146257548
// MI455X (gfx1250) — compile-verified
//
#include <hip/hip_runtime.h>

typedef float v2f __attribute__((ext_vector_type(2)));
typedef float v8f __attribute__((ext_vector_type(8)));

// ---------------- small utility kernels ----------------

__global__ void k_fill(float* __restrict__ p, float v, int n) {
  int i = blockIdx.x * blockDim.x + threadIdx.x;
  if (i < n) p[i] = v;
}

__global__ void k_count_deg(const int* __restrict__ dst, float* __restrict__ deg, int E) {
  int e = blockIdx.x * blockDim.x + threadIdx.x;
  if (e < E) atomicAdd(&deg[dst[e]], 1.0f);
}

__global__ void k_rsqrt_inplace(float* __restrict__ d, int n) {
  int i = blockIdx.x * blockDim.x + threadIdx.x;
  if (i < n) d[i] = rsqrtf(d[i]);  // deg >= 1 always (self-loops folded in)
}

// ---------------- fp32 WMMA GEMM: H[MxC] = X[MxK] @ W[KxC] (+bias) ----------------
// One wave per 16x16 output tile. M % 16 == 0, K % 4 == 0. Ragged C handled by
// clamped B-column loads and guarded stores (branches close before the WMMA, so
// EXEC is all-ones at every matrix op, as the ISA requires).
//
// CDNA5 wave32 layouts (ISA 7.12.2):
//   A 16x4 f32 : lane m=lane&15, h=lane>>4 -> VGPR0=K(2h), VGPR1=K(2h+1)
//   B 4x16 f32 : lane n=lane&15, h=lane>>4 -> VGPR0=K(2h) row, VGPR1=K(2h+1) row
//   C/D 16x16  : VGPR r -> row mtile + 8h + r, col ntile + (lane&15)
__global__ void k_wmma_gemm_f32(const float* __restrict__ X, const float* __restrict__ W,
                                const float* __restrict__ bias, float* __restrict__ H,
                                int K, int C) {
  const int lane  = threadIdx.x & 31;
  const int l15   = lane & 15;
  const int half  = lane >> 4;
  const int mtile = blockIdx.x << 4;
  const int ntile = (blockIdx.y * blockDim.y + threadIdx.y) << 4;

  const int n  = ntile + l15;
  const int nc = (n < C) ? n : (C - 1);  // clamp OOB columns; their D cols are never stored

  const float* xrow = X + (size_t)(mtile + l15) * K + 2 * half;  // A: two consecutive K elems
  const float* wcol = W + (size_t)(2 * half) * C + nc;           // B: two consecutive K rows

  v8f acc = {};
  for (int k0 = 0; k0 < K; k0 += 4) {
    v2f a = *(const v2f*)(xrow + k0);
    v2f b;
    b.x = wcol[(size_t)k0 * C];
    b.y = wcol[(size_t)(k0 + 1) * C];
    acc = __builtin_amdgcn_wmma_f32_16x16x4_f32(false, a, false, b, (short)0, acc,
                                                false, false);
  }

  if (n < C) {
    float bv = bias ? bias[n] : 0.0f;
    float* out = H + (size_t)(mtile + half * 8) * C + n;
#pragma unroll
    for (int r = 0; r < 8; ++r)
      out[(size_t)r * C] = acc[r] + bv;
  }
}

// ---------------- edge-parallel aggregation (L2-resident gather + atomic scatter) ----
// One wave per edge; lanes stride the channel dimension (coalesced gathers).
__global__ void k_edge_agg(const int* __restrict__ src, const int* __restrict__ dst,
                           const float* __restrict__ dinv, const float* __restrict__ H,
                           float* __restrict__ agg, int E, int C) {
  int t = blockIdx.x * blockDim.x + threadIdx.x;
  int e = t >> 5;
  int lane = t & 31;
  if (e >= E) return;
  int s = src[e];
  int d = dst[e];
  float norm = dinv[s] * dinv[d];
  const float* hs = H + (size_t)s * C;
  float* ad = agg + (size_t)d * C;
  for (int c = lane; c < C; c += 32)
    atomicAdd(ad + c, norm * hs[c]);
}

// ---------------- fused self-loop + bias + ReLU ----------------
// agg[i,c] = relu(agg[i,c] + dinv[i]^2 * H[i,c] + bias[c])
__global__ void k_finalize_relu(const float* __restrict__ H, const float* __restrict__ dinv,
                                const float* __restrict__ bias, float* __restrict__ agg,
                                int total, int C) {
  int i = blockIdx.x * blockDim.x + threadIdx.x;
  if (i >= total) return;
  int node = i / C;
  int c = i - node * C;
  float di = dinv[node];
  float v = agg[i] + di * di * H[i] + bias[c];
  agg[i] = v > 0.0f ? v : 0.0f;
}

// ---------------- host orchestration ----------------

extern "C" void kernel_launch(void* const* d_in, const int* in_sizes, int n_in,
                              void* d_out, int out_size, void* d_ws, size_t ws_size,
                              hipStream_t stream) {
  (void)n_in; (void)out_size; (void)ws_size;

  const float* x   = (const float*)d_in[0];
  const int*   ei  = (const int*)d_in[1];   // edge_index [2, E], int32 per harness
  const float* W1  = (const float*)d_in[2];
  const float* b1  = (const float*)d_in[3];
  const float* W2  = (const float*)d_in[4];
  const float* b2  = (const float*)d_in[5];
  const float* Wfc = (const float*)d_in[6];
  const float* bfc = (const float*)d_in[7];
  float* out = (float*)d_out;

  const int IN_C = 128, HID = 128, OUT_C = 64, NCLS = 40;
  const int N = in_sizes[0] / IN_C;   // 100000 (multiple of 16)
  const int E = in_sizes[1] / 2;      // 1600000
  const int* src = ei;
  const int* dst = ei + E;

  // workspace: dinv[N] | bufA[N*128] | bufB[N*128]
  char* ws = (char*)d_ws;
  size_t off0 = ((size_t)N * sizeof(float) + 255) & ~(size_t)255;
  float* dinv = (float*)ws;
  float* bufA = (float*)(ws + off0);
  float* bufB = (float*)(ws + off0 + (size_t)N * HID * sizeof(float));

  const int TB = 256;
  dim3 blk1(TB);

  // 1) degree (self-loop = init 1.0) -> d^{-1/2}
  k_fill<<<dim3((N + TB - 1) / TB), blk1, 0, stream>>>(dinv, 1.0f, N);
  k_count_deg<<<dim3((E + TB - 1) / TB), blk1, 0, stream>>>(dst, dinv, E);
  k_rsqrt_inplace<<<dim3((N + TB - 1) / TB), blk1, 0, stream>>>(dinv, N);

  dim3 gemmBlk(32, 4);  // 4 waves per block, one 16-col tile each
  int mTiles = N / 16;

  // 2) layer 1: H1 = x @ W1 -> bufA
  k_wmma_gemm_f32<<<dim3(mTiles, (HID / 16 + 3) / 4), gemmBlk, 0, stream>>>(
      x, W1, (const float*)nullptr, bufA, IN_C, HID);
  // agg1 = 0 -> bufB
  k_fill<<<dim3((N * HID + TB - 1) / TB), blk1, 0, stream>>>(bufB, 0.0f, N * HID);
  // edge scatter-add
  k_edge_agg<<<dim3((E + 7) / 8), blk1, 0, stream>>>(src, dst, dinv, bufA, bufB, E, HID);
  // + self-loop + b1, ReLU  (bufB := relu(...))
  k_finalize_relu<<<dim3((N * HID + TB - 1) / TB), blk1, 0, stream>>>(
      bufA, dinv, b1, bufB, N * HID, HID);

  // 3) layer 2: H2 = bufB @ W2 -> bufA (reuse)
  k_wmma_gemm_f32<<<dim3(mTiles, (OUT_C / 16 + 3) / 4), gemmBlk, 0, stream>>>(
      bufB, W2, (const float*)nullptr, bufA, HID, OUT_C);
  // agg2 = 0 -> bufB (reuse; prior reads completed in stream order)
  k_fill<<<dim3((N * OUT_C + TB - 1) / TB), blk1, 0, stream>>>(bufB, 0.0f, N * OUT_C);
  k_edge_agg<<<dim3((E + 7) / 8), blk1, 0, stream>>>(src, dst, dinv, bufA, bufB, E, OUT_C);
  k_finalize_relu<<<dim3((N * OUT_C + TB - 1) / TB), blk1, 0, stream>>>(
      bufA, dinv, b2, bufB, N * OUT_C, OUT_C);

  // 4) FC head: out = bufB @ Wfc + bfc  (C=40, ragged tile handled in-kernel)
  k_wmma_gemm_f32<<<dim3(mTiles, ((NCLS + 15) / 16 + 3) / 4), gemmBlk, 0, stream>>>(
      bufB, Wfc, bfc, out, OUT_C, NCLS);
}